// OrthonormalTransform_85220741087551
// MI455X (gfx1250) — compile-verified
//
#include <hip/hip_runtime.h>
#include <hip/hip_bf16.h>

typedef __attribute__((ext_vector_type(2))) float v2f;
typedef __attribute__((ext_vector_type(8))) float v8f;

#define NROT 2016  // 64*63/2

// ---------------------------------------------------------------------------
// Kernel 1: build R' = diag(mus) * (G_2015 ... G_0 @ I).
// One block, 64 threads. Thread j owns column j of the matrix, fully in
// registers (rotation schedule is static -> full unroll -> constant indices).
// ---------------------------------------------------------------------------
__global__ __launch_bounds__(64) void build_R_kernel(
    const float* __restrict__ angles,
    const float* __restrict__ mus,
    float* __restrict__ Rp /* 64x64 row-major */) {
  __shared__ float cs_s[NROT];
  __shared__ float sn_s[NROT];

  const int j = threadIdx.x;  // column owned by this thread

  // Precompute cos/sin for all rotation angles (v_cos_f32 / v_sin_f32).
  for (int i = j; i < NROT; i += 64) {
    float a = angles[i];
    cs_s[i] = __cosf(a);
    sn_s[i] = __sinf(a);
  }
  __syncthreads();

  // Column j of identity.
  float col[64];
#pragma unroll
  for (int r = 0; r < 64; ++r) col[r] = (r == j) ? 1.0f : 0.0f;

  // Apply the 2016 Givens rotations. t,b are compile-time constants after
  // unrolling, so col[] stays entirely in VGPRs.
  int idx = 0;
#pragma unroll
  for (int t = 0; t < 63; ++t) {
#pragma unroll
    for (int b = t + 1; b < 64; ++b) {
      const float c = cs_s[idx];
      const float s = sn_s[idx];
      const float vt = col[t];
      const float vb = col[b];
      col[t] = c * vt - s * vb;
      col[b] = s * vt + c * vb;
      ++idx;
    }
  }

  // R'[r][j] = mus[r] * R[r][j]
#pragma unroll
  for (int r = 0; r < 64; ++r) {
    Rp[r * 64 + j] = mus[r] * col[r];
  }
}

// ---------------------------------------------------------------------------
// Kernel 2: Y = R' @ X using V_WMMA_F32_16X16X4_F32.
// Each wave computes 64-row x 16-col output tiles, grid-striding over column
// tiles. A fragments (R') are register-resident for the whole kernel; R' also
// stays L2-resident (16 KB shared by all waves), so HBM traffic is just
// X reads + Y writes -> memory-bound at ~23.3 TB/s.
//
// WMMA fp32 lane layouts (ISA 7.12.2):
//   A 16x4 (MxK): lane L: M = L&15, K = vgpr + 2*(L>>4)
//   B 4x16 (KxN): vgpr j, lane L: K = j + 2*(L>>4), N = L&15
//   C/D 16x16:    vgpr r, lane L: M = r + 8*(L>>4), N = L&15
// ---------------------------------------------------------------------------
__global__ __launch_bounds__(256) void rgemm_kernel(
    const float* __restrict__ Rp,  // 64x64 row-major
    const float* __restrict__ X,   // 64 x M row-major
    float* __restrict__ Y,         // 64 x M row-major
    int M, int numWaves) {
  const int lane = threadIdx.x & 31;
  const int wave = threadIdx.x >> 5;
  const int gwave = blockIdx.x * (blockDim.x >> 5) + wave;
  const int half = lane >> 4;   // selects K-half for A/B, M-half for C/D
  const int nm = lane & 15;     // N (cols) for B/D, M (rows) for A

  // Load all A fragments once: a[rt][kk] covers rows 16*rt..+15, K 4*kk..+3.
  v2f a[4][16];
#pragma unroll
  for (int rt = 0; rt < 4; ++rt) {
#pragma unroll
    for (int kk = 0; kk < 16; ++kk) {
      const int row = 16 * rt + nm;
      const int k0 = 4 * kk + 2 * half;
      a[rt][kk].x = Rp[row * 64 + k0];
      a[rt][kk].y = Rp[row * 64 + k0 + 1];
    }
  }

  for (int m0 = gwave * 16; m0 < M; m0 += numWaves * 16) {
    // Load all B fragments for this 64x16 X tile (coalesced b32 loads:
    // lanes 0-15 and 16-31 each read 64 contiguous bytes of a row).
    v2f b[16];
#pragma unroll
    for (int kk = 0; kk < 16; ++kk) {
      const float* xp = X + (size_t)(4 * kk + 2 * half) * (size_t)M + m0 + nm;
      b[kk].x = xp[0];
      b[kk].y = xp[(size_t)M];
    }

    v8f acc0 = {}, acc1 = {}, acc2 = {}, acc3 = {};
#pragma unroll
    for (int kk = 0; kk < 16; ++kk) {
      acc0 = __builtin_amdgcn_wmma_f32_16x16x4_f32(false, a[0][kk], false, b[kk],
                                                   (short)0, acc0, false, false);
      acc1 = __builtin_amdgcn_wmma_f32_16x16x4_f32(false, a[1][kk], false, b[kk],
                                                   (short)0, acc1, false, false);
      acc2 = __builtin_amdgcn_wmma_f32_16x16x4_f32(false, a[2][kk], false, b[kk],
                                                   (short)0, acc2, false, false);
      acc3 = __builtin_amdgcn_wmma_f32_16x16x4_f32(false, a[3][kk], false, b[kk],
                                                   (short)0, acc3, false, false);
    }

    // Store D tiles (coalesced 64B segments per half-wave).
    v8f accs[4] = {acc0, acc1, acc2, acc3};
#pragma unroll
    for (int rt = 0; rt < 4; ++rt) {
#pragma unroll
      for (int r = 0; r < 8; ++r) {
        Y[(size_t)(16 * rt + r + 8 * half) * (size_t)M + m0 + nm] = accs[rt][r];
      }
    }
  }
}

extern "C" void kernel_launch(void* const* d_in, const int* in_sizes, int n_in,
                              void* d_out, int out_size, void* d_ws, size_t ws_size,
                              hipStream_t stream) {
  const float* X = (const float*)d_in[0];
  const float* angles = (const float*)d_in[1];
  const float* mus = (const float*)d_in[2];
  float* Y = (float*)d_out;
  float* Rp = (float*)d_ws;  // 64*64 floats = 16 KB scratch

  const int M = in_sizes[0] / 64;

  build_R_kernel<<<1, 64, 0, stream>>>(angles, mus, Rp);

  const int blocks = 1024;           // 8 waves per 256-thread block
  const int numWaves = blocks * 8;   // 8192 waves -> 4 tiles each at M=524288
  rgemm_kernel<<<blocks, 256, 0, stream>>>(Rp, X, Y, M, numWaves);
}